// GAT_85787676771077
// MI455X (gfx1250) — compile-verified
//
#include <hip/hip_runtime.h>
#include <math.h>

// Problem constants (from reference)
#define NN    50000
#define EE    800000
#define EP    850000          // EE + NN self-loops
#define FH    128             // F == HC == 128
#define HEADS 4
#define CH    32
#define GG    64
#define LINW  256
#define OUTC  10
#define NEG_SLOPE 0.2f

typedef __attribute__((ext_vector_type(2))) float v2f;
typedef __attribute__((ext_vector_type(8))) float v8f;

// ---- monotonic float <-> uint encoding for atomic max on f32 ----
__device__ __forceinline__ unsigned enc_f(float f) {
    unsigned u = __float_as_uint(f);
    return (u & 0x80000000u) ? ~u : (u | 0x80000000u);
}
__device__ __forceinline__ float dec_f(unsigned v) {
    return __uint_as_float((v & 0x80000000u) ? (v ^ 0x80000000u) : ~v);
}

__device__ __forceinline__ int esrc(const int* __restrict__ ei, int e) {
    return (e < EE) ? ei[e] : (e - EE);
}
__device__ __forceinline__ int edst(const int* __restrict__ ei, int e) {
    return (e < EE) ? ei[EE + e] : (e - EE);
}

// ============================================================
// GEMM: H = X @ W   (X: [NN x 128] f32, W: [128 x 128] f32)
// One block = 16 rows; 8 waves, each wave owns a 16x16 tile via
// V_WMMA_F32_16X16X4_F32 (32 WMMAs over K=128).
// ============================================================
__global__ void __launch_bounds__(256) gat_gemm_k(
        const float* __restrict__ X, const float* __restrict__ W,
        float* __restrict__ Hout) {
    __shared__ float Alds[16 * 132];          // 132-f pitch: conflict-free column reads
    const int row0 = blockIdx.x * 16;
    const int t = threadIdx.x;

    // Stage the 16x128 A tile (512 float4 loads across 256 threads)
    for (int i = t; i < 16 * 32; i += 256) {
        const int r  = i >> 5;
        const int c4 = i & 31;
        const float4 v = ((const float4*)(X + (size_t)(row0 + r) * FH))[c4];
        float* p = &Alds[r * 132 + c4 * 4];
        p[0] = v.x; p[1] = v.y; p[2] = v.z; p[3] = v.w;
    }
    __syncthreads();

    const int wave = t >> 5;
    const int lane = t & 31;
    const int col0 = wave * 16;
    const int mrow = lane & 15;
    const int ksel = (lane >> 4) << 1;        // 0 for lanes 0-15, 2 for lanes 16-31

    v8f acc = {};
    #pragma unroll
    for (int k = 0; k < FH; k += 4) {
        v2f a, b;
        a[0] = Alds[mrow * 132 + k + ksel];
        a[1] = Alds[mrow * 132 + k + ksel + 1];
        b[0] = W[(size_t)(k + ksel)     * FH + col0 + mrow];
        b[1] = W[(size_t)(k + ksel + 1) * FH + col0 + mrow];
        acc = __builtin_amdgcn_wmma_f32_16x16x4_f32(
                  false, a, false, b, (short)0, acc, false, false);
    }

    // C/D layout: VGPR r -> row r (lanes 0-15) / row r+8 (lanes 16-31), col = lane&15
    const int rbase = row0 + ((lane >> 4) << 3);
    const int c = col0 + mrow;
    #pragma unroll
    for (int r = 0; r < 8; ++r)
        Hout[(size_t)(rbase + r) * FH + c] = acc[r];
}

// ============================================================
// alpha_src / alpha_dst: per-node per-head dot(h, a). One block per
// node, 128 threads; each wave32 == one head -> shfl reduction.
// ============================================================
__global__ void __launch_bounds__(128) gat_alpha_k(
        const float* __restrict__ Hm,
        const float* __restrict__ as_, const float* __restrict__ ad_,
        float* __restrict__ alS, float* __restrict__ alD) {
    const int n = blockIdx.x;
    const int t = threadIdx.x;
    const float hv = Hm[(size_t)n * FH + t];
    float vs = hv * as_[t];
    float vd = hv * ad_[t];
    #pragma unroll
    for (int off = 16; off > 0; off >>= 1) {
        vs += __shfl_xor(vs, off, 32);
        vd += __shfl_xor(vd, off, 32);
    }
    if ((t & 31) == 0) {
        const int head = t >> 5;
        alS[n * HEADS + head] = vs;
        alD[n * HEADS + head] = vd;
    }
}

// zero the accumulator + softmax stats
__global__ void gat_zero_k(float* __restrict__ out, unsigned* __restrict__ mxb,
                           float* __restrict__ den) {
    const int tid = blockIdx.x * blockDim.x + threadIdx.x;
    if (tid < NN * FH) out[tid] = 0.0f;
    if (tid < NN * HEADS) { mxb[tid] = 0u; den[tid] = 0.0f; }
}

// Pass A: logits + segment max (atomic umax on encoded f32)
__global__ void gat_edge_max_k(const int* __restrict__ ei,
                               const float* __restrict__ alS,
                               const float* __restrict__ alD,
                               float* __restrict__ ebuf,
                               unsigned* __restrict__ mxb) {
    const int tid = blockIdx.x * blockDim.x + threadIdx.x;
    if (tid >= EP * HEADS) return;
    const int e = tid >> 2, head = tid & 3;
    const int s = esrc(ei, e), d = edst(ei, e);
    float v = alS[s * HEADS + head] + alD[d * HEADS + head];
    v = (v > 0.0f) ? v : NEG_SLOPE * v;        // leaky relu
    ebuf[tid] = v;
    atomicMax(&mxb[d * HEADS + head], enc_f(v));
}

// Pass B: ex = exp(e - max[dst]); segment sum of ex
__global__ void gat_edge_exp_k(const int* __restrict__ ei,
                               float* __restrict__ ebuf,
                               const unsigned* __restrict__ mxb,
                               float* __restrict__ den) {
    const int tid = blockIdx.x * blockDim.x + threadIdx.x;
    if (tid >= EP * HEADS) return;
    const int e = tid >> 2, head = tid & 3;
    const int d = edst(ei, e);
    const float m = dec_f(mxb[d * HEADS + head]);
    const float ex = __expf(ebuf[tid] - m);
    ebuf[tid] = ex;
    atomicAdd(&den[d * HEADS + head], ex);
}

// Pass C: out[dst] += h[src] * attn. One wave32 per edge; lane owns
// a float4 (head = lane/8). Dominant memory phase -> b128 gathers +
// global_atomic_add_f32 scatters, h stays L2-resident.
__global__ void __launch_bounds__(256) gat_edge_agg_k(
        const int* __restrict__ ei, const float* __restrict__ Hm,
        const float* __restrict__ ebuf, const float* __restrict__ den,
        float* __restrict__ out) {
    const int e = blockIdx.x * 8 + (threadIdx.x >> 5);
    if (e >= EP) return;
    const int l = threadIdx.x & 31;
    const int s = esrc(ei, e), d = edst(ei, e);
    const int head = l >> 3;
    const float attn = ebuf[e * HEADS + head] / den[d * HEADS + head];
    const float4 hv = ((const float4*)(Hm + (size_t)s * FH))[l];
    float* o = out + (size_t)d * FH + 4 * l;
    atomicAdd(o + 0, hv.x * attn);
    atomicAdd(o + 1, hv.y * attn);
    atomicAdd(o + 2, hv.z * attn);
    atomicAdd(o + 3, hv.w * attn);
}

__global__ void gat_bias_relu_k(float* __restrict__ out, const float* __restrict__ b) {
    const int tid = blockIdx.x * blockDim.x + threadIdx.x;
    if (tid >= NN * FH) return;
    const float v = out[tid] + b[tid & (FH - 1)];
    out[tid] = v > 0.0f ? v : 0.0f;
}

__global__ void gat_pool_zero_k(unsigned* __restrict__ pb) {
    const int tid = blockIdx.x * blockDim.x + threadIdx.x;
    if (tid < GG * FH) pb[tid] = 0u;
}

__global__ void gat_pool_max_k(const float* __restrict__ feat,
                               const int* __restrict__ batch,
                               unsigned* __restrict__ pb) {
    const int tid = blockIdx.x * blockDim.x + threadIdx.x;
    if (tid >= NN * FH) return;
    const int n = tid >> 7, c = tid & (FH - 1);
    atomicMax(&pb[batch[n] * FH + c], enc_f(feat[tid]));
}

// pooled @ Wlin + blin, then @ Wout + bout. One block per graph.
__global__ void __launch_bounds__(256) gat_mlp_k(
        const unsigned* __restrict__ pb,
        const float* __restrict__ Wlin, const float* __restrict__ blin,
        const float* __restrict__ Wout, const float* __restrict__ bout,
        float* __restrict__ outp) {
    __shared__ float P[FH];
    __shared__ float Z[LINW];
    const int g = blockIdx.x, t = threadIdx.x;
    if (t < FH) {
        float v = dec_f(pb[g * FH + t]);
        if (!isfinite(v)) v = 0.0f;            // guard empty graphs
        P[t] = v;
    }
    __syncthreads();
    float acc = blin[t];
    #pragma unroll 8
    for (int k = 0; k < FH; ++k) acc += P[k] * Wlin[k * LINW + t];
    Z[t] = acc;
    __syncthreads();
    if (t < OUTC) {
        float o = bout[t];
        #pragma unroll 8
        for (int k = 0; k < LINW; ++k) o += Z[k] * Wout[k * OUTC + t];
        outp[g * OUTC + t] = o;
    }
}

extern "C" void kernel_launch(void* const* d_in, const int* in_sizes, int n_in,
                              void* d_out, int out_size, void* d_ws, size_t ws_size,
                              hipStream_t stream) {
    (void)in_sizes; (void)n_in; (void)out_size; (void)ws_size;
    const float* x     = (const float*)d_in[0];
    const int*   ei    = (const int*)d_in[1];
    const int*   batch = (const int*)d_in[2];
    const float* W[3]  = {(const float*)d_in[3],  (const float*)d_in[7],  (const float*)d_in[11]};
    const float* AS[3] = {(const float*)d_in[4],  (const float*)d_in[8],  (const float*)d_in[12]};
    const float* AD[3] = {(const float*)d_in[5],  (const float*)d_in[9],  (const float*)d_in[13]};
    const float* B[3]  = {(const float*)d_in[6],  (const float*)d_in[10], (const float*)d_in[14]};
    const float* Wlin  = (const float*)d_in[15];
    const float* blin  = (const float*)d_in[16];
    const float* Wout  = (const float*)d_in[17];
    const float* bout  = (const float*)d_in[18];

    char* ws = (char*)d_ws;
    size_t off = 0;
    auto take = [&](size_t bytes) -> char* {
        char* p = ws + off;
        off = (off + bytes + 255) & ~(size_t)255;
        return p;
    };
    float*    bufA = (float*)take((size_t)NN * FH * 4);       // h = feat @ W
    float*    bufB = (float*)take((size_t)NN * FH * 4);       // layer in/out (ping in-place)
    float*    alS  = (float*)take((size_t)NN * HEADS * 4);
    float*    alD  = (float*)take((size_t)NN * HEADS * 4);
    unsigned* mxb  = (unsigned*)take((size_t)NN * HEADS * 4);
    float*    den  = (float*)take((size_t)NN * HEADS * 4);
    float*    ebuf = (float*)take((size_t)EP * HEADS * 4);
    unsigned* pb   = (unsigned*)take((size_t)GG * FH * 4);

    const int ZB = (NN * FH + 255) / 256;
    const int EB = (EP * HEADS + 255) / 256;

    const float* feat = x;
    for (int L = 0; L < 3; ++L) {
        gat_gemm_k<<<NN / 16, 256, 0, stream>>>(feat, W[L], bufA);
        // zero AFTER the gemm: bufB is both this layer's input (L>0) and output
        gat_zero_k<<<ZB, 256, 0, stream>>>(bufB, mxb, den);
        gat_alpha_k<<<NN, 128, 0, stream>>>(bufA, AS[L], AD[L], alS, alD);
        gat_edge_max_k<<<EB, 256, 0, stream>>>(ei, alS, alD, ebuf, mxb);
        gat_edge_exp_k<<<EB, 256, 0, stream>>>(ei, ebuf, mxb, den);
        gat_edge_agg_k<<<(EP + 7) / 8, 256, 0, stream>>>(ei, bufA, ebuf, den, bufB);
        gat_bias_relu_k<<<ZB, 256, 0, stream>>>(bufB, B[L]);
        feat = bufB;
    }
    gat_pool_zero_k<<<(GG * FH + 255) / 256, 256, 0, stream>>>(pb);
    gat_pool_max_k<<<ZB, 256, 0, stream>>>(bufB, batch, pb);
    gat_mlp_k<<<GG, 256, 0, stream>>>(pb, Wlin, blin, Wout, bout, (float*)d_out);
}